// Trajectory_multi_head_80865644249634
// MI455X (gfx1250) — compile-verified
//
#include <hip/hip_runtime.h>
#include <hip/hip_bf16.h>

typedef __attribute__((ext_vector_type(2))) float v2f;
typedef __attribute__((ext_vector_type(8))) float v8f;

// Problem shape from reference setup_inputs()
constexpr int N = 2, T = 7, C = 96, H = 96, W = 96;
constexpr int HW = H * W;
constexpr int HEADS = 4, HD = C / HEADS;   // 4 heads x 24 channels
constexpr int K3 = 3 * C;                  // 288 fused input channels

// ---------------------------------------------------------------------------
// Kernel 1: flow-sample index computation + multi-head max correlation.
// One thread per (n, pixel). Single streaming pass per t accumulates the
// per-head dot products and ||k||^2 simultaneously (both q and k are
// normalized over the FULL channel dim, so score = dot_h / (|q||k|)).
// ---------------------------------------------------------------------------
__global__ __launch_bounds__(256) void traj_corr_kernel(
    const float* __restrict__ curr,    // (n,c,hw)
    const float* __restrict__ indexf,  // (n,t,c,hw)
    const float* __restrict__ locf,    // (n,2t,hw) x/y interleaved per frame
    int*   __restrict__ lin_ws,        // (n,t,hw)  gather index, -1 = invalid
    float* __restrict__ soft_ws,       // (n,hw,4)  winning score per head
    int*   __restrict__ idx_ws)        // (n,hw,4)  winning frame per head
{
    int g = blockIdx.x * blockDim.x + threadIdx.x;
    if (g >= N * HW) return;
    int n = g / HW, p = g % HW;

    const float* q = curr + (size_t)n * C * HW + p;

    // ||q||^2 over full channel dim
    float qn2 = 0.f;
    for (int c = 0; c < C; ++c) {
        float v = q[(size_t)c * HW];
        qn2 += v * v;
    }
    float qinv = 1.0f / fmaxf(sqrtf(qn2), 1e-12f);

    float best[HEADS];
    int   bidx[HEADS];
#pragma unroll
    for (int hh = 0; hh < HEADS; ++hh) { best[hh] = -INFINITY; bidx[hh] = 0; }

    for (int t = 0; t < T; ++t) {
        // align_corners=True nearest sampling of absolute pixel coords
        // simplifies to round-half-even of the raw coordinate.
        float x = locf[((size_t)n * 2 * T + 2 * t + 0) * HW + p];
        float y = locf[((size_t)n * 2 * T + 2 * t + 1) * HW + p];
        float ix = rintf(x), iy = rintf(y);
        bool valid = (ix >= 0.f) && (ix <= (float)(W - 1)) &&
                     (iy >= 0.f) && (iy <= (float)(H - 1));
        int l = valid ? ((int)iy * W + (int)ix) : -1;
        lin_ws[((size_t)n * T + t) * HW + p] = l;

        float sc[HEADS] = {0.f, 0.f, 0.f, 0.f};
        if (l >= 0) {
            const float* kf = indexf + ((size_t)(n * T + t)) * C * HW + l;
            float kn2 = 0.f;
            float dot[HEADS];
#pragma unroll
            for (int hh = 0; hh < HEADS; ++hh) {
                float d = 0.f;
                for (int cc = 0; cc < HD; ++cc) {
                    int c = hh * HD + cc;
                    float kv = kf[(size_t)c * HW];
                    kn2 += kv * kv;
                    d += kv * q[(size_t)c * HW];
                }
                dot[hh] = d;
            }
            float kinv = 1.0f / fmaxf(sqrtf(kn2), 1e-12f);
#pragma unroll
            for (int hh = 0; hh < HEADS; ++hh) sc[hh] = dot[hh] * qinv * kinv;
        }
        // first-max semantics: strictly-greater replaces
#pragma unroll
        for (int hh = 0; hh < HEADS; ++hh) {
            if (sc[hh] > best[hh]) { best[hh] = sc[hh]; bidx[hh] = t; }
        }
    }

#pragma unroll
    for (int hh = 0; hh < HEADS; ++hh) {
        soft_ws[((size_t)n * HW + p) * HEADS + hh] = best[hh];
        idx_ws [((size_t)n * HW + p) * HEADS + hh] = bidx[hh];
    }
}

// ---------------------------------------------------------------------------
// Kernel 2: winning-frame gather + fusion/proj channel-mixing via f32 WMMA.
// Block = 16 pixels, 192 threads = 6 wave32s; wave w owns output rows
// [16w, 16w+16). D(96x16) = fusion_w(96x288) x cat(288x16), soft/bias,
// then D2(96x16) = proj_w(96x96) x mid(96x16) + bias + anchor.
// A-layout (16x4 f32): lane&15 = M row; lanes<16 hold K0/K1, lanes>=16 K2/K3.
// B-layout (4x16 f32): lane&15 = N col; same K split.
// C/D layout: VGPR r = row mrow0 + (lane>>4)*8 + r, col = lane&15.
// ---------------------------------------------------------------------------
__global__ __launch_bounds__(192) void traj_fuse_kernel(
    const float* __restrict__ anchor,  // (n,c,hw)
    const float* __restrict__ f1,      // (n,t,c,hw) sparse set 1
    const float* __restrict__ f2,      // (n,t,c,hw) sparse set 2
    const float* __restrict__ f3,      // (n,t,c,hw) sparse set 3
    const float* __restrict__ fw,      // (c, 3c)   fusion weight
    const float* __restrict__ fbias,   // (c,)
    const float* __restrict__ pw,      // (c, c)    proj weight
    const float* __restrict__ pbias,   // (c,)
    const int*   __restrict__ lin_ws,
    const float* __restrict__ soft_ws,
    const int*   __restrict__ idx_ws,
    float* __restrict__ out)           // (n,c,hw)
{
    __shared__ float cat_lds[K3 * 16];  // 18 KB, row-major [K][16]
    __shared__ float mid_lds[C * 16];   //  6 KB, row-major [C][16]

    const int n   = blockIdx.y;
    const int p0  = blockIdx.x * 16;
    const int tid = threadIdx.x;

    // ---- Phase A: cooperative gather of cat tile (288 x 16) into LDS ----
    for (int i = tid; i < K3 * 16; i += 192) {
        int k = i >> 4, col = i & 15;
        int set = k / C, ch = k - set * C, head = ch / HD;
        int p = p0 + col;
        int tsel = idx_ws[((size_t)n * HW + p) * HEADS + head];
        int l    = lin_ws[((size_t)n * T + tsel) * HW + p];
        const float* sp = (set == 0) ? f1 : (set == 1) ? f2 : f3;
        float v = 0.f;
        if (l >= 0) v = sp[(((size_t)(n * T + tsel)) * C + ch) * HW + l];
        cat_lds[i] = v;
    }
    __syncthreads();

    const int lane  = tid & 31;
    const int wave  = tid >> 5;        // 0..5
    const int mrow0 = wave * 16;       // output-channel tile base
    const int col   = lane & 15;
    const int hi    = lane >> 4;       // K-pair select (0: K0/K1, 1: K2/K3)
    const int p     = p0 + col;
    const int arow  = mrow0 + col;

    // ---- Phase B: fusion matmul, K = 288 in steps of 4 ----
    v8f acc = {};
    for (int k0 = 0; k0 < K3; k0 += 4) {
        int kb = k0 + hi * 2;
        v2f a, b;
        a.x = fw[(size_t)arow * K3 + kb];
        a.y = fw[(size_t)arow * K3 + kb + 1];
        b.x = cat_lds[kb * 16 + col];
        b.y = cat_lds[(kb + 1) * 16 + col];
        acc = __builtin_amdgcn_wmma_f32_16x16x4_f32(
            false, a, false, b, (short)0, acc, false, false);
    }

    // soft scale per head for this pixel
    float s0 = soft_ws[((size_t)n * HW + p) * HEADS + 0];
    float s1 = soft_ws[((size_t)n * HW + p) * HEADS + 1];
    float s2 = soft_ws[((size_t)n * HW + p) * HEADS + 2];
    float s3 = soft_ws[((size_t)n * HW + p) * HEADS + 3];

#pragma unroll
    for (int r = 0; r < 8; ++r) {
        int row  = mrow0 + hi * 8 + r;
        int head = row / HD;
        float sv = (head == 0) ? s0 : (head == 1) ? s1 : (head == 2) ? s2 : s3;
        mid_lds[row * 16 + col] = (acc[r] + fbias[row]) * sv;
    }
    __syncthreads();

    // ---- Phase C: proj matmul, K = 96 in steps of 4 ----
    v8f acc2 = {};
    for (int k0 = 0; k0 < C; k0 += 4) {
        int kb = k0 + hi * 2;
        v2f a, b;
        a.x = pw[(size_t)arow * C + kb];
        a.y = pw[(size_t)arow * C + kb + 1];
        b.x = mid_lds[kb * 16 + col];
        b.y = mid_lds[(kb + 1) * 16 + col];
        acc2 = __builtin_amdgcn_wmma_f32_16x16x4_f32(
            false, a, false, b, (short)0, acc2, false, false);
    }

#pragma unroll
    for (int r = 0; r < 8; ++r) {
        int row = mrow0 + hi * 8 + r;
        size_t o = ((size_t)n * C + row) * HW + p;
        out[o] = acc2[r] + pbias[row] + anchor[o];
    }
}

// ---------------------------------------------------------------------------
extern "C" void kernel_launch(void* const* d_in, const int* in_sizes, int n_in,
                              void* d_out, int out_size, void* d_ws, size_t ws_size,
                              hipStream_t stream) {
    const float* curr   = (const float*)d_in[0];   // curr_feat
    const float* indexf = (const float*)d_in[1];   // index_feat_set_s1
    const float* anchor = (const float*)d_in[2];   // anchor_feat
    const float* f1     = (const float*)d_in[3];   // sparse_feat_set_s1
    const float* f2     = (const float*)d_in[4];   // sparse_feat_set_s2
    const float* f3     = (const float*)d_in[5];   // sparse_feat_set_s3
    const float* locf   = (const float*)d_in[6];   // location_feat
    const float* pw     = (const float*)d_in[7];   // proj_w
    const float* pb     = (const float*)d_in[8];   // proj_b
    const float* fw     = (const float*)d_in[9];   // fusion_w
    const float* fb     = (const float*)d_in[10];  // fusion_b

    // Workspace layout: lin (n,t,hw) i32 | soft (n,hw,4) f32 | idx (n,hw,4) i32
    char* ws = (char*)d_ws;
    int*   lin_ws  = (int*)ws;
    float* soft_ws = (float*)(ws + sizeof(int) * (size_t)N * T * HW);
    int*   idx_ws  = (int*)(ws + sizeof(int) * (size_t)N * T * HW
                               + sizeof(float) * (size_t)N * HW * HEADS);

    traj_corr_kernel<<<(N * HW + 255) / 256, 256, 0, stream>>>(
        curr, indexf, locf, lin_ws, soft_ws, idx_ws);

    dim3 g2(HW / 16, N);
    traj_fuse_kernel<<<g2, 192, 0, stream>>>(
        anchor, f1, f2, f3, fw, fb, pw, pb,
        lin_ws, soft_ws, idx_ws, (float*)d_out);
}